// Yolo_49873160241553
// MI455X (gfx1250) — compile-verified
//
#include <hip/hip_runtime.h>
#include <hip/hip_bf16.h>
#include <math.h>

typedef __attribute__((ext_vector_type(2))) float v2f;
typedef __attribute__((ext_vector_type(8))) float v8f;

#define SDIM 52
#define SS 2704            // 52*52
#define CNUM 80
#define NC 85              // 5 + 80
#define NCH 255            // 3 * 85
#define BATCH 64
#define MBOX 50
#define NBOX (BATCH*MBOX)  // 3200
#define CELL 8.0f          // 416/52
#define MASK_WORDS 85      // ceil(2704/32)

__constant__ float d_aw[9] = {10.f,16.f,33.f,30.f,62.f,59.f,116.f,156.f,373.f};
__constant__ float d_ah[9] = {13.f,30.f,23.f,61.f,45.f,119.f,90.f,198.f,326.f};

// Exact 32-lane f32 sum through the matrix pipe:
// A(16x4) holds {s0,s1} per lane (lanes 0..15 = K0/K1, lanes 16..31 = K2/K3),
// B = ones(4x16)  =>  D[m][n] = rowsum(A,m).  Each lane's 8 D regs cover
// M=0..7 (lanes 0-15) or M=8..15 (lanes 16-31); one xor-16 shuffle finishes.
__device__ __forceinline__ float wave_sum32(float s0, float s1) {
#if __has_builtin(__builtin_amdgcn_wmma_f32_16x16x4_f32)
    v2f a;    a[0] = s0;  a[1] = s1;
    v2f ones; ones[0] = 1.0f; ones[1] = 1.0f;
    v8f cz = {};
    v8f d = __builtin_amdgcn_wmma_f32_16x16x4_f32(
        /*neg_a=*/false, a, /*neg_b=*/false, ones,
        /*c_mod=*/(short)0, cz, /*reuse_a=*/false, /*reuse_b=*/false);
    float part = d[0]+d[1]+d[2]+d[3]+d[4]+d[5]+d[6]+d[7];
    return part + __shfl_xor(part, 16, 32);
#else
    float v = s0 + s1;
    #pragma unroll
    for (int off = 16; off > 0; off >>= 1) v += __shfl_xor(v, off, 32);
    return v;
#endif
}

__device__ __forceinline__ float bce_term(int k, float v, int cls) {
    if (k < 5 || k >= NC) return 0.0f;
    float lab = 1.0f / (1.0f + expf(-v));
    float lg  = fmaxf(logf(lab),    -100.0f);
    float lg1 = fmaxf(log1pf(-lab), -100.0f);
    return ((k - 5) == cls) ? lg : lg1;
}

__global__ void yolo_zero_mask(unsigned int* __restrict__ mask) {
    int i = blockIdx.x * blockDim.x + threadIdx.x;
    if (i < BATCH * 3 * MASK_WORDS) mask[i] = 0u;
}

// One wave (32 lanes) per ground-truth box. 8 waves/block, 400 blocks exactly.
__global__ void yolo_box_kernel(const float* __restrict__ x,
                                const float* __restrict__ geo,
                                const int*   __restrict__ clsv,
                                float*       __restrict__ boxloss,
                                unsigned int* __restrict__ mask) {
    const int wave = (blockIdx.x * blockDim.x + threadIdx.x) >> 5;
    const int lane = threadIdx.x & 31;
    if (wave >= NBOX) return;                    // grid is exact; uniform guard
    const int b = wave / MBOX;

    const float cx = geo[wave*4 + 0];
    const float cy = geo[wave*4 + 1];
    const float w  = geo[wave*4 + 2];
    const float h  = geo[wave*4 + 3];
    const int  cls = clsv[wave];

    // Anchor argmax with the reference's quirky IoU: CM=(min(aw,w)+1)*(min(ah,h)-1)
    float best = -3.4e38f; int bidx = 0;
    #pragma unroll
    for (int i = 0; i < 9; ++i) {
        float aw = d_aw[i], ah = d_ah[i];
        float Aa = (aw + 1.0f) * (ah + 1.0f);
        float Bv = (w  + 1.0f) * (h  + 1.0f);
        float CM = (fminf(aw, w) + 1.0f) * (fminf(ah, h) - 1.0f);
        float iou = CM / (Aa + Bv - CM);
        if (iou > best) { best = iou; bidx = i; }  // strict '>' == first argmax
    }
    const bool valid = (bidx <= 2);
    const int  now   = (bidx < 2) ? bidx : 2;

    int ix = (int)(cx * (1.0f/CELL)); ix = ix < 0 ? 0 : (ix > SDIM-1 ? SDIM-1 : ix);
    int iy = (int)(cy * (1.0f/CELL)); iy = iy < 0 ? 0 : (iy > SDIM-1 ? SDIM-1 : iy);
    const float ax = (cx - (float)ix * CELL) * (1.0f/CELL);
    const float ay = (cy - (float)iy * CELL) * (1.0f/CELL);

    // Gather 85 channel values at (b, now*85+k, ix, iy), lane k strided by 32.
    const float* pv = x + ((size_t)(b * NCH + now * NC)) * SS + (size_t)ix * SDIM + iy;
    const int k0 = lane, k1 = lane + 32, k2 = lane + 64;
    const float va = pv[(size_t)k0 * SS];
    const float vb = pv[(size_t)k1 * SS];
    const float vc = (k2 < NC) ? pv[(size_t)k2 * SS] : 0.0f;

    // Class BCE: per-lane partials packed 2-wide for the WMMA wave reduction.
    const float s0 = bce_term(k0, va, cls) + bce_term(k2, vc, cls);
    const float s1 = bce_term(k1, vb, cls);
    const float bsum = wave_sum32(s0, s1);       // EXEC all ones here
    const float bce = -bsum * (1.0f / (float)CNUM);

    // Broadcast vec[0..4] from lanes 0..4 (first chunk holds k = lane).
    const float vec0 = __shfl(va, 0, 32);
    const float vec1 = __shfl(va, 1, 32);
    const float vec2 = __shfl(va, 2, 32);
    const float vec3 = __shfl(va, 3, 32);
    const float vec4 = __shfl(va, 4, 32);

    const float obj = vec0;
    const float rax = 1.0f / (1.0f + expf(-vec1));
    const float ray = 1.0f / (1.0f + expf(-vec2));
    const float rw  = d_aw[bidx] * expf(vec3);
    const float rh  = d_ah[bidx] * expf(vec4);

    const float p1x = rax*CELL - rw*0.5f, p1y = ray*CELL - rh*0.5f;
    const float p2x = rax*CELL + rw*0.5f, p2y = ray*CELL + rh*0.5f;
    const float g1x = ax*CELL  - w*0.5f,  g1y = ay*CELL  - h*0.5f;
    const float g2x = ax*CELL  + w*0.5f,  g2y = ay*CELL  + h*0.5f;
    const float A1 = (p2x - p1x + 1.0f) * (p2y - p1y + 1.0f);
    const float A2 = (g2x - g1x + 1.0f) * (g2y - g1y + 1.0f);
    const float CMt = (fminf(p2x, g2x) - fmaxf(p1x, g1x) + 1.0f)
                    * (fminf(p2y, g2y) - fmaxf(p1y, g1y + 1.0f));
    const float iou_t = CMt / (A1 + A2 - CMt);

    const float per = 5.0f * (obj - iou_t) * (obj - iou_t) + bce
                    + (rax - ax) * (rax - ax) + (ray - ay) * (ray - ay)
                    + (rw - w) * (rw - w) + (rh - h) * (rh - h);

    if (lane == 0) {
        boxloss[wave] = valid ? per : 0.0f;
        if (valid) {
            const int p = ix * SDIM + iy;
            atomicOr(&mask[(b * 3 + now) * MASK_WORDS + (p >> 5)], 1u << (p & 31));
        }
    }
}

// One block per batch: sum x^2 over non-removed cells of channels {0,85,170}.
__global__ void yolo_noobj_kernel(const float* __restrict__ x,
                                  const unsigned int* __restrict__ mask,
                                  float* __restrict__ contrib) {
    __shared__ float sA[256];
    __shared__ float sB[256];
    const int b = blockIdx.x;
    const int tid = threadIdx.x;
    float ss = 0.0f, cnt = 0.0f;
    for (int idx = tid; idx < 3 * SS; idx += 256) {
        const int c = idx / SS;
        const int p = idx - c * SS;
        const float* addr = x + ((size_t)(b * NCH + c * NC)) * SS + p;
        __builtin_prefetch(addr + 1024, 0, 0);   // global_prefetch_b8, ~4KB ahead
        const float v = *addr;
        const unsigned int mw = mask[(b * 3 + c) * MASK_WORDS + (p >> 5)];
        if (((mw >> (p & 31)) & 1u) == 0u) { ss += v * v; cnt += 1.0f; }
    }
    sA[tid] = ss; sB[tid] = cnt;
    __syncthreads();
    for (int off = 128; off > 0; off >>= 1) {
        if (tid < off) { sA[tid] += sA[tid + off]; sB[tid] += sB[tid + off]; }
        __syncthreads();
    }
    if (tid == 0) contrib[b] = sA[0] / sB[0];
}

// Single-block deterministic final reduction.
__global__ void yolo_final_kernel(const float* __restrict__ boxloss,
                                  const float* __restrict__ contrib,
                                  float* __restrict__ out) {
    __shared__ float sr[256];
    const int tid = threadIdx.x;
    float acc = 0.0f;
    for (int i = tid; i < NBOX;  i += 256) acc += boxloss[i];
    for (int i = tid; i < BATCH; i += 256) acc += 0.5f * contrib[i];
    sr[tid] = acc;
    __syncthreads();
    for (int off = 128; off > 0; off >>= 1) {
        if (tid < off) sr[tid] += sr[tid + off];
        __syncthreads();
    }
    if (tid == 0) out[0] = sr[0];
}

extern "C" void kernel_launch(void* const* d_in, const int* in_sizes, int n_in,
                              void* d_out, int out_size, void* d_ws, size_t ws_size,
                              hipStream_t stream) {
    const float* x    = (const float*)d_in[0];   // (64,255,52,52) f32
    const float* geo  = (const float*)d_in[1];   // (64,50,4) f32
    const int*   clsv = (const int*)  d_in[2];   // (64,50) i32
    float* out = (float*)d_out;

    unsigned int* mask    = (unsigned int*)d_ws;                       // 65280 B
    float*        boxloss = (float*)((char*)d_ws + 65536);             // 12800 B
    float*        contrib = (float*)((char*)d_ws + 65536 + NBOX * 4);  //   256 B

    yolo_zero_mask  <<<64,       256, 0, stream>>>(mask);
    yolo_box_kernel <<<NBOX / 8, 256, 0, stream>>>(x, geo, clsv, boxloss, mask);
    yolo_noobj_kernel<<<BATCH,   256, 0, stream>>>(x, mask, contrib);
    yolo_final_kernel<<<1,       256, 0, stream>>>(boxloss, contrib, out);
}